// CustomRNN_67095979099004
// MI455X (gfx1250) — compile-verified
//
#include <hip/hip_runtime.h>

// ---------------------------------------------------------------------------
// CustomRNN on MI455X (gfx1250): persistent grid-synced recurrent GEMM with
// V_WMMA_F32_16X16X4_F32 (4 independent accumulator chains), W_h2h resident
// in LDS, async GLOBAL->LDS staging of h_t (ASYNCcnt), plus a decoupled
// streaming WMMA kernel for the O=3 readout.
// ---------------------------------------------------------------------------

typedef float v2f __attribute__((ext_vector_type(2)));
typedef float v4f __attribute__((ext_vector_type(4)));
typedef float v8f __attribute__((ext_vector_type(8)));

#define B_    64
#define T_    1024
#define TP1   1025
#define I_    16
#define H_    512
#define O_    3
#define NIN   103
#define NEX   409

#define NWG   32          // persistent workgroups (grid-synced)
#define NTHR  128         // 4 waves (wave32)
#define LDSS  516         // padded K stride (512 + 4) to spread LDS banks
#define LDSB_COLS 64      // W_h2h columns per WG
#define LDSA_ROWS 16      // h rows per WG (one M tile)

// D = A(16x4 f32) x B(4x16 f32) + C(16x16 f32)
__device__ __forceinline__ v8f wmma_f32(v2f a, v2f b, v8f c) {
  return __builtin_amdgcn_wmma_f32_16x16x4_f32(false, a, false, b,
                                               (short)0, c, false, false);
}

// CDNA5 async memory->LDS copy, GVS addressing: 16B per lane, ASYNCcnt.
__device__ __forceinline__ void async_load_b128(int lds_off, unsigned goff,
                                                const void* sbase) {
  asm volatile("global_load_async_to_lds_b128 %0, %1, %2"
               :: "v"(lds_off), "v"(goff), "s"(sbase) : "memory");
}
__device__ __forceinline__ void wait_async0() {
  asm volatile("s_wait_asynccnt 0" ::: "memory");
}

// Grid-wide barrier: monotonic counter (zeroed each launch via memsetAsync).
__device__ __forceinline__ void grid_sync(unsigned* cnt, unsigned target) {
  __syncthreads();
  if (threadIdx.x == 0) {
    __threadfence();  // make this WG's h_new stores device-visible
    __hip_atomic_fetch_add(cnt, 1u, __ATOMIC_RELEASE, __HIP_MEMORY_SCOPE_AGENT);
    while (__hip_atomic_load(cnt, __ATOMIC_ACQUIRE, __HIP_MEMORY_SCOPE_AGENT) < target) {
      __builtin_amdgcn_s_sleep(1);
    }
    __threadfence();  // invalidate stale lines before next step's reads
  }
  __syncthreads();
}

// ---------------------------------------------------------------------------
// Kernel 1: the recurrence.  32 WGs x 4 waves = 128 waves = 4 M-tiles x 32
// N-tiles of the per-step [64x512]x[512x512] f32 GEMM.  Each wave owns one
// 16x16 output tile of h_{t+1}.  h_t lives in the rnn_output region of d_out
// (rnn[:,t,:] == h_t), so the grid barrier is the only cross-step machinery.
// ---------------------------------------------------------------------------
__global__ void __launch_bounds__(NTHR, 1)
rnn_recurrent_kernel(const float* __restrict__ x,       // [B][T][I]
                     const float* __restrict__ noise,   // [B][T][H]
                     const float* __restrict__ W_i2h,   // [H][I]
                     const float* __restrict__ b_i2h,   // [H]
                     const float* __restrict__ W_h2h,   // [H][H]
                     const float* __restrict__ b_h2h,   // [H]
                     const int*   __restrict__ taup,
                     const int*   __restrict__ dtp,
                     float*       __restrict__ rnn,     // [B][1025][H]
                     unsigned*    __restrict__ bar)
{
  extern __shared__ float smem[];
  float* lds_b = smem;                     // [64][516]  W_h2h^T slice (B matrix)
  float* lds_a = smem + LDSB_COLS * LDSS;  // [16][516]  h_t tile (A matrix)

  const int tid  = threadIdx.x;
  const int wave = tid >> 5;
  const int lane = tid & 31;
  const int l16  = lane & 15;
  const int half = lane >> 4;      // K-half select for f32 WMMA A/B layout
  const int koff = half * 2;
  const int wg   = blockIdx.x;
  const int m    = wg & 3;         // M tile (batch rows)
  const int ng   = wg >> 2;        // N group (0..7)
  const int row0 = m * 16;
  const int col0 = ng * 64;
  const int col  = col0 + wave * 16 + l16;   // this lane's output column

  const float alpha  = (float)dtp[0] / (float)taup[0];
  const float nscale = sqrtf(2.0f * alpha) * 0.01f;   // REC_NOISE_STD

  // LDS byte offset of the dynamic-LDS block (static LDS size == 0 here,
  // but derive it properly).
  const unsigned dynbase = __builtin_amdgcn_groupstaticsize();
  const unsigned ldsAbase = dynbase + (unsigned)(LDSB_COLS * LDSS) * 4u;

  // Stage this WG's 64 columns of W_h2h into LDS as B[k][col]: lds_b[c][k].
  for (int i = tid; i < LDSB_COLS * 128; i += NTHR) {
    int c = i >> 7, q = (i & 127) << 2;
    v4f v = *(const v4f*)(W_h2h + (size_t)(col0 + c) * H_ + q);
    *(v4f*)(lds_b + c * LDSS + q) = v;
  }

  // h0 = 0  (rnn[:,0,:])
  for (int i = wg * NTHR + tid; i < B_ * H_; i += NWG * NTHR) {
    int b = i >> 9, h = i & (H_ - 1);
    rnn[(size_t)b * TP1 * H_ + h] = 0.0f;
  }

  // i2h B tile (K=16 -> 4 k-steps) lives in registers; biases fold together:
  // pre = alpha*(x@Wi^T + b_i2h + h@Wh^T + b_h2h) + noise_scaled
  v2f bx0 = *(const v2f*)(W_i2h + col * I_ +  0 + koff);
  v2f bx1 = *(const v2f*)(W_i2h + col * I_ +  4 + koff);
  v2f bx2 = *(const v2f*)(W_i2h + col * I_ +  8 + koff);
  v2f bx3 = *(const v2f*)(W_i2h + col * I_ + 12 + koff);
  const float bsum = b_i2h[col] + b_h2h[col];

  const float* ap = lds_a + l16 * LDSS + koff;
  const float* bp = lds_b + (wave * 16 + l16) * LDSS + koff;
  const size_t xbase = (size_t)(row0 + l16) * T_ * I_ + koff;

  // Per-thread async-staging addresses (t-invariant parts).
  const int   st_r = tid >> 3;                 // 0..15  (row within tile)
  const int   st_q = (tid & 7) << 6;           // 0,64,...,448 (float index)
  const unsigned st_goff = (unsigned)(((row0 + st_r) * TP1) * H_ + st_q) * 4u;
  const unsigned st_lds  = ldsAbase + (unsigned)(st_r * LDSS + st_q) * 4u;

  unsigned target = NWG;
  grid_sync(bar, target);   // h0 + LDS weights ready everywhere

  for (int t = 0; t < T_; ++t) {
    // --- Async-stage A = h_t rows [row0,row0+16) into LDS: each thread DMAs
    // 16 floats (64 B) as 4x b128, no VGPR round trip, tracked by ASYNCcnt.
    {
      const void* sbase = (const void*)(rnn + (size_t)t * H_);
#pragma unroll
      for (int j = 0; j < 4; ++j) {
        async_load_b128((int)(st_lds + 16u * j), st_goff + 16u * j, sbase);
      }
    }

    v8f acc0 = {0,0,0,0,0,0,0,0};
    v8f acc1 = {0,0,0,0,0,0,0,0};
    v8f acc2 = {0,0,0,0,0,0,0,0};
    v8f acc3 = {0,0,0,0,0,0,0,0};

    // --- Overlap with the DMA: i2h contribution (K=16) from global x.
    const float* xp = x + xbase + (size_t)t * I_;
    v2f ax0 = *(const v2f*)(xp + 0);
    v2f ax1 = *(const v2f*)(xp + 4);
    v2f ax2 = *(const v2f*)(xp + 8);
    v2f ax3 = *(const v2f*)(xp + 12);
    acc0 = wmma_f32(ax0, bx0, acc0);
    acc1 = wmma_f32(ax1, bx1, acc1);
    acc2 = wmma_f32(ax2, bx2, acc2);
    acc3 = wmma_f32(ax3, bx3, acc3);

    // --- Also overlap the epilogue's noise loads with the DMA + GEMM.
    float nz[8];
#pragma unroll
    for (int v = 0; v < 8; ++v) {
      int row = row0 + v + 8 * half;
      nz[v] = noise[(size_t)row * T_ * H_ + (size_t)t * H_ + col];
    }

    wait_async0();      // our async LDS writes landed
    __syncthreads();    // everyone's landed

    // --- h2h contribution (K=512), 4 independent accumulator chains.
#pragma unroll 2
    for (int k = 0; k < H_; k += 16) {
      v2f a0 = *(const v2f*)(ap + k);
      v2f b0 = *(const v2f*)(bp + k);
      v2f a1 = *(const v2f*)(ap + k + 4);
      v2f b1 = *(const v2f*)(bp + k + 4);
      v2f a2 = *(const v2f*)(ap + k + 8);
      v2f b2 = *(const v2f*)(bp + k + 8);
      v2f a3 = *(const v2f*)(ap + k + 12);
      v2f b3 = *(const v2f*)(bp + k + 12);
      acc0 = wmma_f32(a0, b0, acc0);
      acc1 = wmma_f32(a1, b1, acc1);
      acc2 = wmma_f32(a2, b2, acc2);
      acc3 = wmma_f32(a3, b3, acc3);
    }
    acc0 += acc1;
    acc2 += acc3;
    acc0 += acc2;

    // Epilogue: pre = alpha*(acc + bias) + scaled noise; h_new = relu(pre).
    // C layout: VGPR v -> row v (lanes 0-15) / row v+8 (lanes 16-31).
#pragma unroll
    for (int v = 0; v < 8; ++v) {
      int row = row0 + v + 8 * half;
      float pre = alpha * (acc0[v] + bsum) + nz[v] * nscale;
      float hnw = fmaxf(pre, 0.0f);
      rnn[(size_t)row * TP1 * H_ + (size_t)(t + 1) * H_ + col] = hnw;
    }

    target += NWG;
    grid_sync(bar, target);
  }
}

// ---------------------------------------------------------------------------
// Kernel 2: network_output[b,idx,:] = state @ W_h2o^T + b_h2o (idx>=1),
// 0 at idx==0.  rnn_output is a flat [65600][512] row matrix, so tiles of 16
// consecutive (b,idx) rows feed WMMA directly; W_h2o^T is zero-padded to
// 512x16 in LDS (cols 0..2 live, rows <NIN zero).  Pure streaming: 134 MB
// read once from HBM (~6 us at 23.3 TB/s).
// ---------------------------------------------------------------------------
#define K2_BLOCKS 64
#define K2_THREADS 256
#define K2_WAVES 8
#define GROWS (B_ * TP1)        // 65600
#define GTILES (GROWS / 16)     // 4100

__global__ void __launch_bounds__(K2_THREADS, 1)
out_proj_kernel(const float* __restrict__ rnn,     // [B*1025][H]
                const float* __restrict__ W_h2o,   // [O][NEX]
                const float* __restrict__ b_h2o,   // [O]
                float*       __restrict__ net)     // [B*1025][O]
{
  __shared__ float bo[16 * LDSS];
  const int tid  = threadIdx.x;
  const int wave = tid >> 5;
  const int lane = tid & 31;
  const int l16  = lane & 15;
  const int half = lane >> 4;
  const int koff = half * 2;

  for (int i = tid; i < 16 * H_; i += K2_THREADS) {
    int c = i >> 9, k = i & (H_ - 1);
    float v = 0.0f;
    if (c < O_ && k >= NIN) v = W_h2o[c * NEX + (k - NIN)];
    bo[c * LDSS + k] = v;
  }
  __syncthreads();

  const float* bp   = bo + l16 * LDSS + koff;
  const float  bias = (l16 < O_) ? b_h2o[l16] : 0.0f;
  const int nwaves  = K2_BLOCKS * K2_WAVES;

  for (int tile = blockIdx.x * K2_WAVES + wave; tile < GTILES; tile += nwaves) {
    const float* apb = rnn + (size_t)tile * 16 * H_ + (size_t)l16 * H_ + koff;
    v8f acc0 = {0,0,0,0,0,0,0,0};
    v8f acc1 = {0,0,0,0,0,0,0,0};
    v8f acc2 = {0,0,0,0,0,0,0,0};
    v8f acc3 = {0,0,0,0,0,0,0,0};
#pragma unroll 2
    for (int k = 0; k < H_; k += 16) {
      v2f a0 = *(const v2f*)(apb + k);
      v2f b0 = *(const v2f*)(bp + k);
      v2f a1 = *(const v2f*)(apb + k + 4);
      v2f b1 = *(const v2f*)(bp + k + 4);
      v2f a2 = *(const v2f*)(apb + k + 8);
      v2f b2 = *(const v2f*)(bp + k + 8);
      v2f a3 = *(const v2f*)(apb + k + 12);
      v2f b3 = *(const v2f*)(bp + k + 12);
      acc0 = wmma_f32(a0, b0, acc0);
      acc1 = wmma_f32(a1, b1, acc1);
      acc2 = wmma_f32(a2, b2, acc2);
      acc3 = wmma_f32(a3, b3, acc3);
    }
    acc0 += acc1;
    acc2 += acc3;
    acc0 += acc2;

    if (l16 < O_) {          // stores only; WMMAs above ran with full EXEC
#pragma unroll
      for (int v = 0; v < 8; ++v) {
        int g   = tile * 16 + v + 8 * half;
        int idx = g % TP1;
        float val = (idx == 0) ? 0.0f : (acc0[v] + bias);
        net[(size_t)g * O_ + l16] = val;
      }
    }
  }
}

// ---------------------------------------------------------------------------
extern "C" void kernel_launch(void* const* d_in, const int* in_sizes, int n_in,
                              void* d_out, int out_size, void* d_ws, size_t ws_size,
                              hipStream_t stream) {
  (void)in_sizes; (void)n_in; (void)out_size; (void)ws_size;
  const float* x     = (const float*)d_in[0];
  const float* noise = (const float*)d_in[1];
  const float* W_i2h = (const float*)d_in[2];
  const float* b_i2h = (const float*)d_in[3];
  const float* W_h2h = (const float*)d_in[4];
  const float* b_h2h = (const float*)d_in[5];
  const float* W_h2o = (const float*)d_in[6];
  const float* b_h2o = (const float*)d_in[7];
  const int*   tau   = (const int*)d_in[8];
  const int*   dt    = (const int*)d_in[9];

  float* net = (float*)d_out;                         // [64][1025][3]
  float* rnn = net + (size_t)B_ * TP1 * O_;           // [64][1025][512]
  unsigned* bar = (unsigned*)d_ws;

  // Barrier counter must start at 0 every call (d_ws is not re-poisoned
  // between replays); async memset is graph-capturable.
  hipMemsetAsync(d_ws, 0, 256, stream);

  const size_t lds1 = (size_t)(LDSB_COLS + LDSA_ROWS) * LDSS * sizeof(float);
  rnn_recurrent_kernel<<<dim3(NWG), dim3(NTHR), lds1, stream>>>(
      x, noise, W_i2h, b_i2h, W_h2h, b_h2h, tau, dt, rnn, bar);
  out_proj_kernel<<<dim3(K2_BLOCKS), dim3(K2_THREADS), 0, stream>>>(
      rnn, W_h2o, b_h2o, net);
}